// NGCF_6536940224900
// MI455X (gfx1250) — compile-verified
//
#include <hip/hip_runtime.h>

#define N_USERS  200000
#define N_ITEMS  200000
#define N_TOT    400000
#define D        64
#define NNZ      12800000
#define N_LAYERS 3
#define OUT_COLS 256      // D * (N_LAYERS + 1)
#define TSTR     66       // padded LDS row stride (66 floats = 264 B, 8B aligned)

typedef float v2f __attribute__((ext_vector_type(2)));
typedef float v8f __attribute__((ext_vector_type(8)));

// ---------------- init: out[:, 0:64] = concat(user_emb, item_emb) ----------------
__global__ void ngcf_init(const float* __restrict__ ue, const float* __restrict__ ie,
                          float* __restrict__ out) {
    long t = (long)blockIdx.x * blockDim.x + threadIdx.x;
    if (t >= (long)N_TOT * D) return;
    int n = (int)(t / D), c = (int)(t % D);
    float v = (n < N_USERS) ? ue[(long)n * D + c] : ie[(long)(n - N_USERS) * D + c];
    out[(long)n * OUT_COLS + c] = v;
}

// ---------------- zero fill ----------------
__global__ void ngcf_zero(float* __restrict__ p, long n) {
    long t = (long)blockIdx.x * blockDim.x + threadIdx.x;
    if (t < n) p[t] = 0.0f;
}

// ---------------- SpMM: side = A @ ego (COO, 16 threads / edge, float4) ----------------
__global__ void ngcf_spmm(const int* __restrict__ rows, const int* __restrict__ cols,
                          const float* __restrict__ vals,
                          const float* __restrict__ ego,   // row stride OUT_COLS
                          float* __restrict__ side) {      // row stride D
    long t = (long)blockIdx.x * blockDim.x + threadIdx.x;
    if (t >= (long)NNZ * 16) return;
    int e = (int)(t >> 4);
    int q = (int)(t & 15);
    int r = rows[e], c = cols[e];
    float v = vals[e];
    const float4 g = *(const float4*)(ego + (long)c * OUT_COLS + q * 4);
    float* dst = side + (long)r * D + q * 4;
    atomicAdd(dst + 0, g.x * v);
    atomicAdd(dst + 1, g.y * v);
    atomicAdd(dst + 2, g.z * v);
    atomicAdd(dst + 3, g.w * v);
}

// ---------------- dense: ego_out = leaky(side@Wgc^T + bgc) + leaky((ego*side)@Wbi^T + bbi)
// 128 threads = 4 waves; each wave owns a 16-row tile; fp32 WMMA 16x16x4.
__global__ __launch_bounds__(128)
void ngcf_dense(const float* __restrict__ side,      // [N, 64]
                const float* __restrict__ ego_in,    // row stride OUT_COLS
                const float* __restrict__ wgc, const float* __restrict__ bgc,
                const float* __restrict__ wbi, const float* __restrict__ bbi,
                float* __restrict__ ego_out) {       // row stride OUT_COLS
    __shared__ float sWg[64 * TSTR];
    __shared__ float sWb[64 * TSTR];
    __shared__ float sS[4][16 * TSTR];
    __shared__ float sE[4][16 * TSTR];

    const int tid  = threadIdx.x;
    const int wave = tid >> 5;
    const int lane = tid & 31;
    const int m    = lane & 15;     // row-in-tile (A) / col-in-tile (B,C,D)
    const int h    = lane >> 4;     // half-wave select

    // stage weights (row-major W[j][k]) into padded LDS
    for (int idx = tid; idx < 64 * 64; idx += 128) {
        int j = idx >> 6, k = idx & 63;
        sWg[j * TSTR + k] = wgc[idx];
        sWb[j * TSTR + k] = wbi[idx];
    }
    // stage this wave's 16x64 side & ego tiles
    const int row0 = blockIdx.x * 64 + wave * 16;
    for (int idx = lane; idx < 16 * 64; idx += 32) {
        int tm = idx >> 6, c = idx & 63;
        sS[wave][tm * TSTR + c] = side[(long)(row0 + tm) * D + c];
        sE[wave][tm * TSTR + c] = ego_in[(long)(row0 + tm) * OUT_COLS + c];
    }
    __syncthreads();

    const float* As = &sS[wave][m * TSTR];
    const float* Ae = &sE[wave][m * TSTR];

    #pragma unroll
    for (int jb = 0; jb < 4; ++jb) {
        const float* Wg = sWg + (jb * 16 + m) * TSTR;
        const float* Wb = sWb + (jb * 16 + m) * TSTR;
        v8f acc0 = {};   // GC branch
        v8f acc1 = {};   // Bi branch
        #pragma unroll
        for (int kc = 0; kc < 16; ++kc) {
            int off = kc * 4 + 2 * h;            // K = 4*kc + 2h + {0,1}
            float s0 = As[off], s1 = As[off + 1];
            v2f as; as.x = s0;            as.y = s1;
            v2f ae; ae.x = s0 * Ae[off];  ae.y = s1 * Ae[off + 1];
            v2f bg; bg.x = Wg[off];       bg.y = Wg[off + 1];
            v2f bb; bb.x = Wb[off];       bb.y = Wb[off + 1];
            acc0 = __builtin_amdgcn_wmma_f32_16x16x4_f32(false, as, false, bg,
                                                         (short)0, acc0, false, false);
            acc1 = __builtin_amdgcn_wmma_f32_16x16x4_f32(false, ae, false, bb,
                                                         (short)0, acc1, false, false);
        }
        float biasg = bgc[jb * 16 + m];
        float biasb = bbi[jb * 16 + m];
        #pragma unroll
        for (int r = 0; r < 8; ++r) {
            float x0 = acc0[r] + biasg; x0 = (x0 > 0.0f) ? x0 : 0.01f * x0;
            float x1 = acc1[r] + biasb; x1 = (x1 > 0.0f) ? x1 : 0.01f * x1;
            int row = row0 + r + h * 8;
            ego_out[(long)row * OUT_COLS + jb * 16 + m] = x0 + x1;
        }
    }
}

// ---------------- final: L2-normalize layer blocks 1..3 of each row ----------------
__global__ __launch_bounds__(192)
void ngcf_norm(float* __restrict__ out) {
    __shared__ float red[192];
    const int n = blockIdx.x;
    const int t = threadIdx.x;
    const int l = t >> 6;       // 0..2 -> layer block l+1
    const int c = t & 63;
    float* p = out + (long)n * OUT_COLS + (l + 1) * 64 + c;
    float v = *p;
    red[t] = v * v;
    __syncthreads();
    for (int s = 32; s > 0; s >>= 1) {
        if (c < s) red[t] += red[t + s];
        __syncthreads();
    }
    float nrm = fmaxf(sqrtf(red[l * 64]), 1e-12f);
    *p = v / nrm;
}

extern "C" void kernel_launch(void* const* d_in, const int* in_sizes, int n_in,
                              void* d_out, int out_size, void* d_ws, size_t ws_size,
                              hipStream_t stream) {
    const int*   rows = (const int*)d_in[0];
    const int*   cols = (const int*)d_in[1];
    const float* vals = (const float*)d_in[2];
    const float* ue   = (const float*)d_in[3];
    const float* ie   = (const float*)d_in[4];
    const float* gcw  = (const float*)d_in[5];
    const float* gcb  = (const float*)d_in[6];
    const float* biw  = (const float*)d_in[7];
    const float* bib  = (const float*)d_in[8];
    float* out  = (float*)d_out;
    float* side = (float*)d_ws;          // N_TOT * D floats (~102 MB)

    {
        long total = (long)N_TOT * D;
        ngcf_init<<<(int)((total + 255) / 256), 256, 0, stream>>>(ue, ie, out);
    }
    for (int l = 0; l < N_LAYERS; ++l) {
        long total = (long)N_TOT * D;
        ngcf_zero<<<(int)((total + 255) / 256), 256, 0, stream>>>(side, total);
        long tspmm = (long)NNZ * 16;
        ngcf_spmm<<<(int)((tspmm + 255) / 256), 256, 0, stream>>>(
            rows, cols, vals, out + l * D, side);
        ngcf_dense<<<N_TOT / 64, 128, 0, stream>>>(
            side, out + l * D,
            gcw + (long)l * D * D, gcb + (long)l * D,
            biw + (long)l * D * D, bib + (long)l * D,
            out + (l + 1) * D);
    }
    ngcf_norm<<<N_TOT, 192, 0, stream>>>(out);
}